// HashEncoding_22428319220119
// MI455X (gfx1250) — compile-verified
//
#include <hip/hip_runtime.h>
#include <stdint.h>

// -----------------------------------------------------------------------------
// Multiresolution hash-grid encoding (Instant-NGP style) for MI455X / gfx1250.
// Gather-bound workload: no matmul structure -> WMMA inapplicable.
// CDNA5-specific paths used:
//   * global_load_async_to_lds_b128 + s_wait_asynccnt (stage g16 table in LDS)
//   * ds_load/ds_store for LDS gathers + coalescing transpose
//   * non-temporal (th:NT) loads for x and stores for the 240MB output so the
//     ~43MB of tables stay resident in the 192MB L2.
// -----------------------------------------------------------------------------

typedef float v2f __attribute__((ext_vector_type(2)));

#define NLEV        15
#define HASH_MASK   ((1u << 19) - 1u)
#define BBOX_F      2.0f
#define PRECOND_F   10.0f
#define BLOCK       256

struct TablePtrs { const float* t[NLEV]; };

// Dense grid level: reference flips axes, so dim0 = z, dim1 = y, dim2 = x.
// Corners are clamped to [0, gs-1]; weights use unclamped fractional t.
__device__ __forceinline__ void grid_level(const float* __restrict__ tab, const int gs,
                                           float x0, float x1, float x2,
                                           float* __restrict__ o)
{
    const float lz = x2 * (float)gs - 0.5f;
    const float ly = x1 * (float)gs - 0.5f;
    const float lx = x0 * (float)gs - 0.5f;
    const float bz = floorf(lz), by = floorf(ly), bx = floorf(lx);
    const float tz = lz - bz, ty = ly - by, tx = lx - bx;
    const int   iz = (int)bz, iy = (int)by, ix = (int)bx;
    const int zi[2] = { min(max(iz,     0), gs - 1), min(max(iz + 1, 0), gs - 1) };
    const int yi[2] = { min(max(iy,     0), gs - 1), min(max(iy + 1, 0), gs - 1) };
    const int xi[2] = { min(max(ix,     0), gs - 1), min(max(ix + 1, 0), gs - 1) };
    const float wz[2] = { 1.0f - tz, tz };
    const float wy[2] = { 1.0f - ty, ty };
    const float wx[2] = { 1.0f - tx, tx };
    float a0 = 0.0f, a1 = 0.0f;
#pragma unroll
    for (int c = 0; c < 8; ++c) {
        const int oz = (c >> 2) & 1, oy = (c >> 1) & 1, ox = c & 1;
        const int idx = ((zi[oz] * gs + yi[oy]) * gs + xi[ox]) * 2;
        const v2f v = *(const v2f*)(tab + idx);
        const float w = wz[oz] * wy[oy] * wx[ox];
        a0 = fmaf(w, v.x, a0);
        a1 = fmaf(w, v.y, a1);
    }
    o[0] = a0 * PRECOND_F;
    o[1] = a1 * PRECOND_F;
}

// Hashed level: axes NOT flipped; offset bit2 -> x, bit1 -> y, bit0 -> z.
// idx = (x*1 ^ y*2654435761 ^ z*805459861) & (2^19 - 1)
__device__ __forceinline__ void hash_level(const float* __restrict__ tab, const int gs,
                                           float x0, float x1, float x2,
                                           float* __restrict__ o)
{
    const float lx = x0 * (float)gs - 0.5f;
    const float ly = x1 * (float)gs - 0.5f;
    const float lz = x2 * (float)gs - 0.5f;
    const float bx = floorf(lx), by = floorf(ly), bz = floorf(lz);
    const float tx = lx - bx, ty = ly - by, tz = lz - bz;
    const int   ix = (int)bx, iy = (int)by, iz = (int)bz;
    const uint32_t hx[2] = { (uint32_t)ix,                    (uint32_t)(ix + 1) };
    const uint32_t hy[2] = { (uint32_t)iy * 2654435761u,      (uint32_t)(iy + 1) * 2654435761u };
    const uint32_t hz[2] = { (uint32_t)iz * 805459861u,       (uint32_t)(iz + 1) * 805459861u };
    const float wx[2] = { 1.0f - tx, tx };
    const float wy[2] = { 1.0f - ty, ty };
    const float wz[2] = { 1.0f - tz, tz };
    float a0 = 0.0f, a1 = 0.0f;
#pragma unroll
    for (int c = 0; c < 8; ++c) {
        const int ox = (c >> 2) & 1, oy = (c >> 1) & 1, oz = c & 1;
        const uint32_t idx = ((hx[ox] ^ hy[oy] ^ hz[oz]) & HASH_MASK) * 2u;
        const v2f v = *(const v2f*)(tab + idx);
        const float w = wx[ox] * wy[oy] * wz[oz];
        a0 = fmaf(w, v.x, a0);
        a1 = fmaf(w, v.y, a1);
    }
    o[0] = a0 * PRECOND_F;
    o[1] = a1 * PRECOND_F;
}

__global__ void __launch_bounds__(BLOCK)
hash_encode_kernel(const float* __restrict__ xin, TablePtrs tp,
                   float* __restrict__ out, int N)
{
    __shared__ __align__(16) float s_g16[16 * 16 * 16 * 2];  // 32 KiB
    __shared__ __align__(16) float s_out[BLOCK * 30];        // 30 KiB

    const int tid = threadIdx.x;

    // ---- CDNA5 async global->LDS staging of the level-0 grid (32 KiB) -------
    // 8 iterations x 256 threads x 16B = 32 KiB, tracked on ASYNCcnt.
    {
        const uint64_t gbase = (uint64_t)(uintptr_t)(const void*)tp.t[0];
        const uint32_t lbase = (uint32_t)(uintptr_t)(void*)&s_g16[0]; // low 32b of generic = LDS offset
#pragma unroll
        for (int i = 0; i < 8; ++i) {
            const uint32_t boff  = (uint32_t)(tid + i * BLOCK) * 16u;
            const uint32_t loff  = lbase + boff;
            const uint64_t gaddr = gbase + boff;
            asm volatile("global_load_async_to_lds_b128 %0, %1, off"
                         :: "v"(loff), "v"(gaddr) : "memory");
        }
        asm volatile("s_wait_asynccnt 0" ::: "memory");
    }
    __syncthreads();

    const int n = blockIdx.x * BLOCK + tid;
    float feats[30];

    if (n < N) {
        const float px = __builtin_nontemporal_load(&xin[3 * n + 0]);
        const float py = __builtin_nontemporal_load(&xin[3 * n + 1]);
        const float pz = __builtin_nontemporal_load(&xin[3 * n + 2]);
        const float x0 = (px + BBOX_F) * (0.5f / BBOX_F);
        const float x1 = (py + BBOX_F) * (0.5f / BBOX_F);
        const float x2 = (pz + BBOX_F) * (0.5f / BBOX_F);

        grid_level(s_g16,  16, x0, x1, x2, &feats[0]);   // from LDS (ds_load_b64)
        grid_level(tp.t[1], 23, x0, x1, x2, &feats[2]);
        grid_level(tp.t[2], 32, x0, x1, x2, &feats[4]);
        grid_level(tp.t[3], 45, x0, x1, x2, &feats[6]);
        grid_level(tp.t[4], 64, x0, x1, x2, &feats[8]);

        const int HGS[10] = {91, 128, 181, 256, 362, 512, 724, 1024, 1448, 2048};
#pragma unroll
        for (int l = 0; l < 10; ++l)
            hash_level(tp.t[5 + l], HGS[l], x0, x1, x2, &feats[10 + 2 * l]);
    } else {
#pragma unroll
        for (int j = 0; j < 30; ++j) feats[j] = 0.0f;
    }

    // ---- stage to LDS, then coalesced non-temporal writeback ----------------
    // s_out layout matches global: point-major rows of 30 floats.
#pragma unroll
    for (int j = 0; j < 30; ++j) s_out[tid * 30 + j] = feats[j];
    __syncthreads();

    const v2f* s2 = (const v2f*)s_out;
    v2f* o2 = (v2f*)out + (size_t)blockIdx.x * (BLOCK * 15);
    const long long base2  = (long long)blockIdx.x * (BLOCK * 15);
    const long long total2 = (long long)N * 15;
#pragma unroll
    for (int j = 0; j < 15; ++j) {
        const int i = tid + j * BLOCK;
        if (base2 + i < total2)
            __builtin_nontemporal_store(s2[i], &o2[i]);  // th:NT, keep L2 for tables
    }
}

extern "C" void kernel_launch(void* const* d_in, const int* in_sizes, int n_in,
                              void* d_out, int out_size, void* d_ws, size_t ws_size,
                              hipStream_t stream)
{
    const float* x = (const float*)d_in[0];
    TablePtrs tp;
    for (int i = 0; i < NLEV; ++i) tp.t[i] = (const float*)d_in[1 + i];
    float* out = (float*)d_out;

    const int N = in_sizes[0] / 3;
    const int blocks = (N + BLOCK - 1) / BLOCK;
    hash_encode_kernel<<<blocks, BLOCK, 0, stream>>>(x, tp, out, N);
}